// PRank_41351945126167
// MI455X (gfx1250) — compile-verified
//
#include <hip/hip_runtime.h>

typedef __attribute__((ext_vector_type(2))) float v2f;
typedef __attribute__((ext_vector_type(4))) float v4f;
typedef __attribute__((ext_vector_type(8))) float v8f;

#define WORDS 1000000
#define D 64
#define K 4
#define B 8192
#define NBLK 16
#define ROWS_PER_BLK (B / NBLK)   // 512 rows per block
#define WS_STRIDE 80              // per-block partial: [0]=cnt, [1..4]=btau, [8..71]=wsum

// ---------------------------------------------------------------------------
// Kernel 1: dots via V_WMMA_F32_16X16X4_F32, per-row PRank logic, deterministic
// per-block reductions into ws.
// ---------------------------------------------------------------------------
__global__ __launch_bounds__(256) void prank_compute(
    const float* __restrict__ in_embed,
    const float* __restrict__ in_bias,
    const int*   __restrict__ context_id,
    const int*   __restrict__ target_ids,
    const int*   __restrict__ labels,
    float*       __restrict__ ws)
{
    __shared__ float s_ctx[D];
    __shared__ float s_bias[K];
    __shared__ float s_dots[ROWS_PER_BLK];
    __shared__ float s_taus[ROWS_PER_BLK];
    __shared__ int   s_id[ROWS_PER_BLK];
    __shared__ float s_red[256];
    __shared__ float s_w[4][D];

    const int tid = threadIdx.x;
    const int ctx = context_id[0];
    if (tid < D) s_ctx[tid]  = in_embed[(size_t)ctx * D + tid];
    if (tid < K) s_bias[tid] = in_bias[(size_t)ctx * K + tid];
    __syncthreads();

    const int blk  = blockIdx.x;
    const int row0 = blk * ROWS_PER_BLK;
    const int wave = tid >> 5;
    const int lane = tid & 31;
    const int m    = lane & 15;        // M row slot within 16x16 tile
    const int half = lane >> 4;        // lane half selects K pair
    const int koff = half << 1;        // K = koff + v  (v = 0,1)  [ISA 16x4 f32 A layout]

    // ---- Phase 1: 16-row tiles of dots = tgt_emb @ ctx via WMMA f32 16x16x4.
    // B fragment broadcasts ctx across all 16 columns, so every D column holds
    // the dot; lanes m==0 (lanes 0 and 16) scatter the 8 accumulator VGPRs
    // (M = r and M = 8+r per the C/D layout) to LDS.
    for (int t = 0; t < ROWS_PER_BLK / (8 * 16); ++t) {   // 4 tiles per wave
        const int rl = wave * 64 + t * 16 + m;            // local row 0..511
        const int id = target_ids[row0 + rl];
        if (half == 0) s_id[rl] = id;
        const float* erow = in_embed + (size_t)id * D;
        v8f acc = {};
        #pragma unroll
        for (int c = 0; c < 16; ++c) {                    // K = 64 in 16 steps
            v2f a = *(const v2f*)(erow + c * 4 + koff);
            v2f b = *(const v2f*)(&s_ctx[c * 4 + koff]);
            acc = __builtin_amdgcn_wmma_f32_16x16x4_f32(
                false, a, false, b, (short)0, acc, false, false);
        }
        if (m == 0) {
            const int base = wave * 64 + t * 16 + half * 8;
            #pragma unroll
            for (int r = 0; r < 8; ++r) s_dots[base + r] = acc[r];
        }
    }
    __syncthreads();

    // ---- Phase 2: per-row predicted label, tau, accuracy count.
    float cnt = 0.f;
    float btau[K] = {0.f, 0.f, 0.f, 0.f};
    for (int rr = 0; rr < ROWS_PER_BLK / 256; ++rr) {
        const int   rl  = rr * 256 + tid;
        const float dot = s_dots[rl];
        const int   lab = labels[row0 + rl];
        int   p    = K + 1;
        float taus = 0.f;
        #pragma unroll
        for (int k = K - 1; k >= 0; --k) {
            const float db = dot - s_bias[k];
            if (db <= 0.f) p = k + 1;                 // ends at first nonpos k
            const float judge = (k < lab) ? db : -db; // db * yt
            const float tauk  = (judge > 0.f) ? 0.f : (float)lab;
            btau[k] += tauk;
            taus    += tauk;
        }
        if (p == lab) cnt += 1.f;
        s_taus[rl] = taus;
    }
    __syncthreads();

    // ---- Phase 3: weight_update partial: sum_r taus[r] * emb[id[r]][j].
    const int j   = tid & 63;
    const int seg = tid >> 6;        // 4 segments of 128 rows
    float wsum = 0.f;
    for (int r = seg * 128; r < seg * 128 + 128; ++r) {
        const float ts = s_taus[r];
        if (ts != 0.f)
            wsum += ts * in_embed[(size_t)s_id[r] * D + j];
    }
    s_w[seg][j] = wsum;

    // ---- Deterministic block reductions (fixed-pairing trees).
    s_red[tid] = cnt;
    __syncthreads();
    for (int s = 128; s > 0; s >>= 1) {
        if (tid < s) s_red[tid] += s_red[tid + s];
        __syncthreads();
    }
    if (tid == 0) ws[blk * WS_STRIDE + 0] = s_red[0];

    for (int k = 0; k < K; ++k) {
        __syncthreads();
        s_red[tid] = btau[k];
        __syncthreads();
        for (int s = 128; s > 0; s >>= 1) {
            if (tid < s) s_red[tid] += s_red[tid + s];
            __syncthreads();
        }
        if (tid == 0) ws[blk * WS_STRIDE + 1 + k] = s_red[0];
    }
    __syncthreads();
    if (tid < D)
        ws[blk * WS_STRIDE + 8 + tid] =
            s_w[0][tid] + s_w[1][tid] + s_w[2][tid] + s_w[3][tid];
}

// ---------------------------------------------------------------------------
// Kernel 2: streaming copy of embed (64M) + bias (4M) floats to out+1.
// 272 MB in + 272 MB out > 192 MB L2 -> non-temporal on both sides.
// Dest is out+1 (4-mod-16 aligned): NT float4 loads, NT scalar stores.
// ---------------------------------------------------------------------------
__global__ __launch_bounds__(256) void prank_copy(
    const float* __restrict__ in_embed,
    const float* __restrict__ in_bias,
    float*       __restrict__ out)
{
    const size_t EV = (size_t)WORDS * D / 4;   // 16M vec4 (embed)
    const size_t BV = (size_t)WORDS * K / 4;   // 1M vec4 (bias)
    const size_t stride = (size_t)gridDim.x * blockDim.x;
    for (size_t i = (size_t)blockIdx.x * blockDim.x + threadIdx.x;
         i < EV + BV; i += stride) {
        const float* src;
        float* dst;
        if (i < EV) {
            src = in_embed + i * 4;
            dst = out + 1 + i * 4;
        } else {
            const size_t k = i - EV;
            src = in_bias + k * 4;
            dst = out + 1 + (size_t)WORDS * D + k * 4;
        }
        v4f v = __builtin_nontemporal_load((const v4f*)src);
        __builtin_nontemporal_store(v[0], dst + 0);
        __builtin_nontemporal_store(v[1], dst + 1);
        __builtin_nontemporal_store(v[2], dst + 2);
        __builtin_nontemporal_store(v[3], dst + 3);
    }
}

// ---------------------------------------------------------------------------
// Kernel 3: sum the 16 block partials in fixed order, write acc, apply the
// context-row update on top of the copied tables.
// ---------------------------------------------------------------------------
__global__ void prank_finalize(const int* __restrict__ context_id,
                               const float* __restrict__ ws,
                               float* __restrict__ out)
{
    const int j   = threadIdx.x;           // 64 threads
    const int ctx = context_id[0];
    const float inv = 1.0f / (float)B;

    float wsum = 0.f;
    for (int b = 0; b < NBLK; ++b) wsum += ws[b * WS_STRIDE + 8 + j];
    out[1 + (size_t)ctx * D + j] += wsum * inv;

    if (j == 0) {
        float cnt = 0.f;
        for (int b = 0; b < NBLK; ++b) cnt += ws[b * WS_STRIDE];
        out[0] = cnt * inv;
    }
    if (j < K) {
        float bs = 0.f;
        for (int b = 0; b < NBLK; ++b) bs += ws[b * WS_STRIDE + 1 + j];
        out[1 + (size_t)WORDS * D + (size_t)ctx * K + j] -= bs * inv;
    }
}

extern "C" void kernel_launch(void* const* d_in, const int* in_sizes, int n_in,
                              void* d_out, int out_size, void* d_ws, size_t ws_size,
                              hipStream_t stream) {
    const float* in_embed   = (const float*)d_in[0];
    const float* in_bias    = (const float*)d_in[1];
    const int*   context_id = (const int*)d_in[2];
    const int*   target_ids = (const int*)d_in[3];
    const int*   labels     = (const int*)d_in[4];
    float* out = (float*)d_out;
    float* ws  = (float*)d_ws;

    prank_compute<<<NBLK, 256, 0, stream>>>(in_embed, in_bias, context_id,
                                            target_ids, labels, ws);
    prank_copy<<<8192, 256, 0, stream>>>(in_embed, in_bias, out);
    prank_finalize<<<1, 64, 0, stream>>>(context_id, ws, out);
}